// HeteroGCN_69492570849588
// MI455X (gfx1250) — compile-verified
//
#include <hip/hip_runtime.h>

typedef float v2f __attribute__((ext_vector_type(2)));
typedef float v8f __attribute__((ext_vector_type(8)));

#define N_USER  100000
#define N_ITEM  100000
#define NEDGE   800000
#define IN_USER 256
#define IN_ITEM 128
#define HID     128
#define OUTF    64

// ---------------------------------------------------------------------------
// utility kernels
// ---------------------------------------------------------------------------
__global__ void zero4_kernel(float4* __restrict__ p, int n4) {
    int i = blockIdx.x * blockDim.x + threadIdx.x;
    int stride = gridDim.x * blockDim.x;
    const float4 z = {0.f, 0.f, 0.f, 0.f};
    for (; i < n4; i += stride) p[i] = z;
}

__global__ void count_kernel(const int* __restrict__ idx, float* __restrict__ deg, int n) {
    int i = blockIdx.x * blockDim.x + threadIdx.x;
    if (i < n) atomicAdd(&deg[idx[i]], 1.0f);
}

__global__ void inv_kernel(float* __restrict__ deg, int n) {
    int i = blockIdx.x * blockDim.x + threadIdx.x;
    if (i < n) deg[i] = 1.0f / sqrtf(fmaxf(deg[i], 1.0f));
}

// Wt[n*K + k] = W[k*N + n]  (weights are tiny: <= 256*128 floats)
__global__ void transpose_kernel(const float* __restrict__ W, float* __restrict__ Wt,
                                 int K, int N) {
    int i = blockIdx.x * blockDim.x + threadIdx.x;
    if (i >= K * N) return;
    int k = i / N, n = i - k * N;
    Wt[n * K + k] = W[i];
}

// ---------------------------------------------------------------------------
// C[M,N] = (A[M,K] @ W[K,N]) * scale[row]   via V_WMMA_F32_16X16X4_F32
// W supplied TRANSPOSED (Wt[N,K]) so the B fragment is a contiguous float2.
// One wave computes NT adjacent 16x16 tiles (A fragment reused NT times).
// M%16==0, K%4==0, N%(16*NT)==0 -> no guards -> EXEC all ones for every WMMA.
//
// A frag (16x4 f32, 2 VGPR/lane): lane l<16 : A[row0+l][ka=k+0], [ka+1]
//                                 lane l>=16: A[row0+l-16][ka=k+2], [ka+3]
// B frag (4x16 f32): lane half h supplies K = k+2h, k+2h+1 at N = lane%16,
//                    i.e. Wt[col*K + ka], Wt[col*K + ka + 1]  (contiguous).
// C/D frag (16x16 f32, 8 VGPR): VGPR i -> M = (lane/16)*8 + i, N = lane%16.
// ---------------------------------------------------------------------------
template <int K, int N, int NT>
__global__ __launch_bounds__(32 * (N / (16 * NT)))
void gemm_rowscale(const float* __restrict__ A, const float* __restrict__ Wt,
                   const float* __restrict__ scale, float* __restrict__ C) {
    const int lane = threadIdx.x;        // 0..31
    const int half = lane >> 4;          // 0 or 1
    const int l16  = lane & 15;
    const int row0 = blockIdx.x * 16;
    const int colBase = threadIdx.y * (16 * NT);

    const float* __restrict__ arow = A + (row0 + l16) * K;

    v8f acc[NT];
#pragma unroll
    for (int t = 0; t < NT; ++t) acc[t] = (v8f){};

#pragma unroll 4
    for (int k = 0; k < K; k += 4) {
        const int ka = k + 2 * half;
        v2f a = *(const v2f*)(arow + ka);
#pragma unroll
        for (int t = 0; t < NT; ++t) {
            const int col = colBase + t * 16 + l16;
            v2f b = *(const v2f*)(Wt + col * K + ka);
            acc[t] = __builtin_amdgcn_wmma_f32_16x16x4_f32(
                /*neg_a=*/false, a, /*neg_b=*/false, b,
                /*c_mod=*/(short)0, acc[t], /*reuse_a=*/false, /*reuse_b=*/false);
        }
    }

#pragma unroll
    for (int i = 0; i < 8; ++i) {
        const int r = row0 + half * 8 + i;
        const float s = scale[r];
#pragma unroll
        for (int t = 0; t < NT; ++t)
            C[r * N + colBase + t * 16 + l16] = acc[t][i] * s;
    }
}

// ---------------------------------------------------------------------------
// agg[dst[e]] += xw[src[e]]   (F floats per edge, float4 chunks, fp32 atomics)
// consecutive threads = consecutive chunks of the same edge (coalesced gather)
// ---------------------------------------------------------------------------
template <int F>
__global__ void scatter_add(const float* __restrict__ xw,
                            const int* __restrict__ src,
                            const int* __restrict__ dst,
                            float* __restrict__ agg, int nedge) {
    const int chunks = F / 4;
    long long t = (long long)blockIdx.x * blockDim.x + threadIdx.x;
    if (t >= (long long)nedge * chunks) return;
    const int e = (int)(t / chunks);
    const int c = (int)(t % chunks) * 4;
    const int s = src[e];
    const int d = dst[e];
    const float4 v = *(const float4*)(xw + s * F + c);
    float* p = agg + d * F + c;
    atomicAdd(p + 0, v.x);
    atomicAdd(p + 1, v.y);
    atomicAdd(p + 2, v.z);
    atomicAdd(p + 3, v.w);
}

// ---------------------------------------------------------------------------
// out = [relu]( agg * inv[row] + bias[col] )
// ---------------------------------------------------------------------------
__global__ void combine1(const float* __restrict__ agg, const float* __restrict__ inv,
                         const float* __restrict__ bias, float* __restrict__ out,
                         int nnodes, int F, int do_relu) {
    int i = blockIdx.x * blockDim.x + threadIdx.x;
    if (i >= nnodes * F) return;
    const int r = i / F;
    const int j = i - r * F;
    float v = agg[i] * inv[r] + bias[j];
    out[i] = do_relu ? fmaxf(v, 0.0f) : v;
}

// out = [relu]( aggx*invx[row] + bx[col] + aggy*invy[row] + by[col] )
__global__ void combine2(const float* __restrict__ aggx, const float* __restrict__ invx,
                         const float* __restrict__ bx,
                         const float* __restrict__ aggy, const float* __restrict__ invy,
                         const float* __restrict__ by,
                         float* __restrict__ out, int nnodes, int F, int do_relu) {
    int i = blockIdx.x * blockDim.x + threadIdx.x;
    if (i >= nnodes * F) return;
    const int r = i / F;
    const int j = i - r * F;
    float v = aggx[i] * invx[r] + bx[j] + aggy[i] * invy[r] + by[j];
    out[i] = do_relu ? fmaxf(v, 0.0f) : v;
}

// ---------------------------------------------------------------------------
extern "C" void kernel_launch(void* const* d_in, const int* in_sizes, int n_in,
                              void* d_out, int out_size, void* d_ws, size_t ws_size,
                              hipStream_t stream) {
    const float* feat_user = (const float*)d_in[0];   // [100000,256]
    const float* feat_item = (const float*)d_in[1];   // [100000,128]
    const int* rates_src = (const int*)d_in[2];
    const int* rates_dst = (const int*)d_in[3];
    const int* rated_src = (const int*)d_in[4];
    const int* rated_dst = (const int*)d_in[5];
    const int* fol_src   = (const int*)d_in[6];
    const int* fol_dst   = (const int*)d_in[7];
    const float* W1_rates = (const float*)d_in[8];   const float* b1_rates = (const float*)d_in[9];
    const float* W1_rated = (const float*)d_in[10];  const float* b1_rated = (const float*)d_in[11];
    const float* W1_fol   = (const float*)d_in[12];  const float* b1_fol   = (const float*)d_in[13];
    const float* W2_rates = (const float*)d_in[14];  const float* b2_rates = (const float*)d_in[15];
    const float* W2_rated = (const float*)d_in[16];  const float* b2_rated = (const float*)d_in[17];
    const float* W2_fol   = (const float*)d_in[18];  const float* b2_fol   = (const float*)d_in[19];

    float* ws = (float*)d_ws;
    // degree / inv-sqrt arrays (computed once, valid for both layers)
    float* inv_rates_src = ws;               // over users  (deg_out of "rates")
    float* inv_rates_dst = ws + 100000;      // over items  (deg_in  of "rates")
    float* inv_rated_src = ws + 200000;      // over items
    float* inv_rated_dst = ws + 300000;      // over users
    float* inv_fol_src   = ws + 400000;      // over users
    float* inv_fol_dst   = ws + 500000;      // over users
    float* xw    = ws + 600000;              // 100000*128 reusable GEMM output
    float* agg_a = xw    + 12800000;         // 100000*128
    float* agg_b = agg_a + 12800000;         // 100000*128
    float* agg_c = agg_b + 12800000;         // 100000*128
    float* h_user = agg_c + 12800000;        // 100000*128
    float* h_item = h_user + 12800000;       // 100000*128
    float* wt     = h_item + 12800000;       // up to 256*128 transposed weights

    float* out_user = (float*)d_out;                    // [100000,64]
    float* out_item = out_user + N_USER * OUTF;         // [100000,64]

    const int EB = (NEDGE + 255) / 256;

    // ---- degrees -> inv sqrt (shared by both layers) ----
    zero4_kernel<<<256, 256, 0, stream>>>((float4*)ws, 600000 / 4);
    count_kernel<<<EB, 256, 0, stream>>>(rates_src, inv_rates_src, NEDGE);
    count_kernel<<<EB, 256, 0, stream>>>(rates_dst, inv_rates_dst, NEDGE);
    count_kernel<<<EB, 256, 0, stream>>>(rated_src, inv_rated_src, NEDGE);
    count_kernel<<<EB, 256, 0, stream>>>(rated_dst, inv_rated_dst, NEDGE);
    count_kernel<<<EB, 256, 0, stream>>>(fol_src,   inv_fol_src,   NEDGE);
    count_kernel<<<EB, 256, 0, stream>>>(fol_dst,   inv_fol_dst,   NEDGE);
    inv_kernel<<<(600000 + 255) / 256, 256, 0, stream>>>(ws, 600000);

    // ---- layer 1 (HID = 128) ----
    zero4_kernel<<<4096, 256, 0, stream>>>((float4*)agg_a, (3 * 12800000) / 4);

    transpose_kernel<<<(IN_USER * HID + 255) / 256, 256, 0, stream>>>(W1_rates, wt, IN_USER, HID);
    gemm_rowscale<IN_USER, HID, 2><<<dim3(N_USER / 16), dim3(32, HID / 32), 0, stream>>>(
        feat_user, wt, inv_rates_src, xw);
    scatter_add<HID><<<NEDGE * (HID / 4) / 256, 256, 0, stream>>>(
        xw, rates_src, rates_dst, agg_a, NEDGE);

    transpose_kernel<<<(IN_ITEM * HID + 255) / 256, 256, 0, stream>>>(W1_rated, wt, IN_ITEM, HID);
    gemm_rowscale<IN_ITEM, HID, 2><<<dim3(N_ITEM / 16), dim3(32, HID / 32), 0, stream>>>(
        feat_item, wt, inv_rated_src, xw);
    scatter_add<HID><<<NEDGE * (HID / 4) / 256, 256, 0, stream>>>(
        xw, rated_src, rated_dst, agg_b, NEDGE);

    transpose_kernel<<<(IN_USER * HID + 255) / 256, 256, 0, stream>>>(W1_fol, wt, IN_USER, HID);
    gemm_rowscale<IN_USER, HID, 2><<<dim3(N_USER / 16), dim3(32, HID / 32), 0, stream>>>(
        feat_user, wt, inv_fol_src, xw);
    scatter_add<HID><<<NEDGE * (HID / 4) / 256, 256, 0, stream>>>(
        xw, fol_src, fol_dst, agg_c, NEDGE);

    combine1<<<(N_ITEM * HID + 255) / 256, 256, 0, stream>>>(
        agg_a, inv_rates_dst, b1_rates, h_item, N_ITEM, HID, 1);
    combine2<<<(N_USER * HID + 255) / 256, 256, 0, stream>>>(
        agg_b, inv_rated_dst, b1_rated, agg_c, inv_fol_dst, b1_fol, h_user, N_USER, HID, 1);

    // ---- layer 2 (OUT = 64), reuse agg region narrower ----
    float* agg2_a = agg_a;                 // 100000*64
    float* agg2_b = agg_a + 6400000;       // 100000*64
    float* agg2_c = agg_a + 12800000;      // 100000*64
    zero4_kernel<<<4096, 256, 0, stream>>>((float4*)agg_a, (3 * 6400000) / 4);

    transpose_kernel<<<(HID * OUTF + 255) / 256, 256, 0, stream>>>(W2_rates, wt, HID, OUTF);
    gemm_rowscale<HID, OUTF, 2><<<dim3(N_USER / 16), dim3(32, OUTF / 32), 0, stream>>>(
        h_user, wt, inv_rates_src, xw);
    scatter_add<OUTF><<<NEDGE * (OUTF / 4) / 256, 256, 0, stream>>>(
        xw, rates_src, rates_dst, agg2_a, NEDGE);

    transpose_kernel<<<(HID * OUTF + 255) / 256, 256, 0, stream>>>(W2_rated, wt, HID, OUTF);
    gemm_rowscale<HID, OUTF, 2><<<dim3(N_ITEM / 16), dim3(32, OUTF / 32), 0, stream>>>(
        h_item, wt, inv_rated_src, xw);
    scatter_add<OUTF><<<NEDGE * (OUTF / 4) / 256, 256, 0, stream>>>(
        xw, rated_src, rated_dst, agg2_b, NEDGE);

    transpose_kernel<<<(HID * OUTF + 255) / 256, 256, 0, stream>>>(W2_fol, wt, HID, OUTF);
    gemm_rowscale<HID, OUTF, 2><<<dim3(N_USER / 16), dim3(32, OUTF / 32), 0, stream>>>(
        h_user, wt, inv_fol_src, xw);
    scatter_add<OUTF><<<NEDGE * (OUTF / 4) / 256, 256, 0, stream>>>(
        xw, fol_src, fol_dst, agg2_c, NEDGE);

    combine1<<<(N_ITEM * OUTF + 255) / 256, 256, 0, stream>>>(
        agg2_a, inv_rates_dst, b2_rates, out_item, N_ITEM, OUTF, 0);
    combine2<<<(N_USER * OUTF + 255) / 256, 256, 0, stream>>>(
        agg2_b, inv_rated_dst, b2_rated, agg2_c, inv_fol_dst, b2_fol, out_user, N_USER, OUTF, 0);
}